// GATConvConcat_55997783605394
// MI455X (gfx1250) — compile-verified
//
#include <hip/hip_runtime.h>
#include <hip/hip_bf16.h>

#define N_NODES 50000
#define N_EDGES 800000
#define IN_FEATS 256
#define HEADS 4
#define OUT_FEATS 32
#define HF (HEADS * OUT_FEATS)   // 128
#define NEG_SLOPE 0.2f

typedef __attribute__((ext_vector_type(2))) float v2f;
typedef __attribute__((ext_vector_type(8))) float v8f;

// ---------------------------------------------------------------------------
// helpers
// ---------------------------------------------------------------------------
__device__ __forceinline__ float leaky(float x) {
    return (x >= 0.0f) ? x : NEG_SLOPE * x;
}

// float atomic max via monotone int encoding (lowers to global_atomic_max_i32
// / global_atomic_min_u32 which are native L2 atomics on CDNA5)
__device__ __forceinline__ void atomicMaxFloat(float* addr, float val) {
    if (val >= 0.0f) {
        atomicMax((int*)addr, __float_as_int(val));
    } else {
        atomicMin((unsigned int*)addr, __float_as_uint(val));
    }
}

// ---------------------------------------------------------------------------
// Kernel 1: feat_hf = feat @ W_fc^T  via V_WMMA_F32_16X16X4_F32
//   feat : [M, 256] row-major, W_fc : [128, 256] row-major (so B = W_fc^T)
//   One wave computes a 16-row x 128-col strip: 8 tiles of 16x16, K loop of 64.
// ---------------------------------------------------------------------------
__global__ void __launch_bounds__(256)
gat_gemm_wmma(const float* __restrict__ feat,
              const float* __restrict__ Wfc,
              float* __restrict__ out, int M) {
#if __has_builtin(__builtin_amdgcn_wmma_f32_16x16x4_f32)
    const int lane  = threadIdx.x & 31;
    const int wave  = threadIdx.x >> 5;
    const int strip = blockIdx.x * 8 + wave;          // 16-row strip index
    const int nstrip = M >> 4;                        // M % 16 == 0
    if (strip >= nstrip) return;                      // uniform per-wave exit

    const int m0    = strip << 4;
    const int lrow  = lane & 15;                      // M (for A) / N (for B)
    const int koff  = (lane < 16) ? 0 : 2;            // K sub-offset per half

    v8f acc[8];
#pragma unroll
    for (int t = 0; t < 8; ++t) acc[t] = (v8f)(0.0f);

    const float* arow = feat + (size_t)(m0 + lrow) * IN_FEATS + koff;

    for (int k0 = 0; k0 < IN_FEATS; k0 += 4) {
        // A fragment: 16x4 tile of feat (lane L<16 -> K=k0,k0+1 ; L>=16 -> K=k0+2,k0+3)
        v2f a = *(const v2f*)(arow + k0);
#pragma unroll
        for (int t = 0; t < 8; ++t) {
            // B fragment: 4x16 tile of W_fc^T -> rows of W_fc
            const float* brow = Wfc + (size_t)(t * 16 + lrow) * IN_FEATS + k0 + koff;
            v2f b = *(const v2f*)brow;
            acc[t] = __builtin_amdgcn_wmma_f32_16x16x4_f32(
                false, a, false, b, (short)0, acc[t], false, false);
        }
    }

    // C/D layout: VGPR r -> row (m0 + r + (lane<16?0:8)), col = t*16 + (lane&15)
    const int rbase = m0 + ((lane < 16) ? 0 : 8);
#pragma unroll
    for (int t = 0; t < 8; ++t) {
#pragma unroll
        for (int r = 0; r < 8; ++r) {
            out[(size_t)(rbase + r) * HF + t * 16 + lrow] = acc[t][r];
        }
    }
#endif
}

// ---------------------------------------------------------------------------
// Kernel 2: el[n,h] = sum_f feat_hf[n,h,f]*attn_l[h,f]  (and er with attn_r)
//   One wave per node; lane owns cols [4l, 4l+3] (all within head l>>3);
//   per-head reduction over 8 lanes via shfl_xor.
// ---------------------------------------------------------------------------
__global__ void __launch_bounds__(256)
gat_eler(const float* __restrict__ fh,
         const float* __restrict__ attn_l,
         const float* __restrict__ attn_r,
         float* __restrict__ el, float* __restrict__ er, int N) {
    const int lane = threadIdx.x & 31;
    const int node = (blockIdx.x * blockDim.x + threadIdx.x) >> 5;
    if (node >= N) return;

    const float4 f4 = ((const float4*)(fh + (size_t)node * HF))[lane];
    const float4 al = ((const float4*)attn_l)[lane];
    const float4 ar = ((const float4*)attn_r)[lane];

    float pl = f4.x * al.x + f4.y * al.y + f4.z * al.z + f4.w * al.w;
    float pr = f4.x * ar.x + f4.y * ar.y + f4.z * ar.z + f4.w * ar.w;
#pragma unroll
    for (int m = 1; m < 8; m <<= 1) {
        pl += __shfl_xor(pl, m, 32);
        pr += __shfl_xor(pr, m, 32);
    }
    if ((lane & 7) == 0) {
        const int h = lane >> 3;
        el[node * HEADS + h] = pl;
        er[node * HEADS + h] = pr;
    }
}

// ---------------------------------------------------------------------------
// Kernel 3: init m = -inf, s = 0, out(=numerator) = 0
// ---------------------------------------------------------------------------
__global__ void __launch_bounds__(256)
gat_init(float* __restrict__ m, float* __restrict__ s,
         float* __restrict__ out, int total) {
    const int i = blockIdx.x * blockDim.x + threadIdx.x;
    if (i >= total) return;
    m[i]   = __int_as_float(0xFF800000);  // -inf
    s[i]   = 0.0f;
    out[i] = 0.0f;
}

// ---------------------------------------------------------------------------
// Kernel 4: segment max. One warp per edge (grid-stride); lane owns hf = 4l..4l+3.
// ---------------------------------------------------------------------------
__global__ void __launch_bounds__(256)
gat_edge_max(const float* __restrict__ ew,
             const int* __restrict__ src, const int* __restrict__ dst,
             const float* __restrict__ el, const float* __restrict__ er,
             const float* __restrict__ Wedge,
             float* __restrict__ m, int E) {
    const int lane  = threadIdx.x & 31;
    const int warp  = (blockIdx.x * blockDim.x + threadIdx.x) >> 5;
    const int nwarp = (gridDim.x * blockDim.x) >> 5;
    const float4 we = ((const float4*)Wedge)[lane];   // W_edge[4l..4l+3]
    const int h = lane >> 3;

    for (int e = warp; e < E; e += nwarp) {
        const int   sn = src[e];
        const int   dn = dst[e];
        const float w  = ew[e];
        const float base = el[sn * HEADS + h] + er[dn * HEADS + h];
        float* mrow = m + (size_t)dn * HF + lane * 4;
        atomicMaxFloat(mrow + 0, leaky(base + w * we.x));
        atomicMaxFloat(mrow + 1, leaky(base + w * we.y));
        atomicMaxFloat(mrow + 2, leaky(base + w * we.z));
        atomicMaxFloat(mrow + 3, leaky(base + w * we.w));
    }
}

// ---------------------------------------------------------------------------
// Kernel 5: accumulate denominator s[dst] += exp(v-m) and
//           numerator out[dst] += feat_hf[src] * exp(v-m)
// ---------------------------------------------------------------------------
__global__ void __launch_bounds__(256)
gat_edge_acc(const float* __restrict__ ew,
             const int* __restrict__ src, const int* __restrict__ dst,
             const float* __restrict__ el, const float* __restrict__ er,
             const float* __restrict__ Wedge,
             const float* __restrict__ m, const float* __restrict__ fh,
             float* __restrict__ s, float* __restrict__ num, int E) {
    const int lane  = threadIdx.x & 31;
    const int warp  = (blockIdx.x * blockDim.x + threadIdx.x) >> 5;
    const int nwarp = (gridDim.x * blockDim.x) >> 5;
    const float4 we = ((const float4*)Wedge)[lane];
    const int h = lane >> 3;

    for (int e = warp; e < E; e += nwarp) {
        const int   sn = src[e];
        const int   dn = dst[e];
        const float w  = ew[e];
        const float base = el[sn * HEADS + h] + er[dn * HEADS + h];

        const float4 mv = ((const float4*)(m  + (size_t)dn * HF))[lane];
        const float4 fv = ((const float4*)(fh + (size_t)sn * HF))[lane];

        const float ex0 = __expf(leaky(base + w * we.x) - mv.x);
        const float ex1 = __expf(leaky(base + w * we.y) - mv.y);
        const float ex2 = __expf(leaky(base + w * we.z) - mv.z);
        const float ex3 = __expf(leaky(base + w * we.w) - mv.w);

        float* srow = s   + (size_t)dn * HF + lane * 4;
        float* nrow = num + (size_t)dn * HF + lane * 4;
        atomicAdd(srow + 0, ex0);
        atomicAdd(srow + 1, ex1);
        atomicAdd(srow + 2, ex2);
        atomicAdd(srow + 3, ex3);
        atomicAdd(nrow + 0, fv.x * ex0);
        atomicAdd(nrow + 1, fv.y * ex1);
        atomicAdd(nrow + 2, fv.z * ex2);
        atomicAdd(nrow + 3, fv.w * ex3);
    }
}

// ---------------------------------------------------------------------------
// Kernel 6: out = (s != 0 ? out/s : 0) + bias
// ---------------------------------------------------------------------------
__global__ void __launch_bounds__(256)
gat_finalize(float* __restrict__ out, const float* __restrict__ s,
             const float* __restrict__ bias, int total) {
    const int i = blockIdx.x * blockDim.x + threadIdx.x;
    if (i >= total) return;
    const float sv = s[i];
    const float r  = (sv != 0.0f) ? (out[i] / sv) : 0.0f;
    out[i] = r + bias[i & (HF - 1)];
}

// ---------------------------------------------------------------------------
// launch
// ---------------------------------------------------------------------------
extern "C" void kernel_launch(void* const* d_in, const int* in_sizes, int n_in,
                              void* d_out, int out_size, void* d_ws, size_t ws_size,
                              hipStream_t stream) {
    const float* feat   = (const float*)d_in[0];
    const float* ew     = (const float*)d_in[1];
    const int*   src    = (const int*)  d_in[2];
    const int*   dst    = (const int*)  d_in[3];
    const float* Wfc    = (const float*)d_in[4];
    const float* Wedge  = (const float*)d_in[5];
    const float* attn_l = (const float*)d_in[6];
    const float* attn_r = (const float*)d_in[7];
    const float* bias   = (const float*)d_in[8];

    const int N = in_sizes[0] / IN_FEATS;   // 50000
    const int E = in_sizes[1];              // 800000
    const int total = N * HF;

    // workspace layout (floats)
    float* fh = (float*)d_ws;               // N*HF
    float* m  = fh + (size_t)total;         // N*HF
    float* s  = m  + (size_t)total;         // N*HF
    float* el = s  + (size_t)total;         // N*HEADS
    float* er = el + (size_t)N * HEADS;     // N*HEADS
    float* num = (float*)d_out;             // numerator accumulated in-place

    // 1. projection GEMM (WMMA f32)
    {
        const int nstrip = N >> 4;
        const int blocks = (nstrip + 7) / 8;
        gat_gemm_wmma<<<blocks, 256, 0, stream>>>(feat, Wfc, fh, N);
    }
    // 2. el / er
    {
        const int blocks = (N + 7) / 8;     // 8 waves per block, 1 node per wave
        gat_eler<<<blocks, 256, 0, stream>>>(fh, attn_l, attn_r, el, er, N);
    }
    // 3. init m/s/num
    {
        const int blocks = (total + 255) / 256;
        gat_init<<<blocks, 256, 0, stream>>>(m, s, num, total);
    }
    // 4. segment max
    {
        gat_edge_max<<<2048, 256, 0, stream>>>(ew, src, dst, el, er, Wedge, m, E);
    }
    // 5. segment sums (denominator + numerator)
    {
        gat_edge_acc<<<2048, 256, 0, stream>>>(ew, src, dst, el, er, Wedge,
                                               m, fh, s, num, E);
    }
    // 6. finalize
    {
        const int blocks = (total + 255) / 256;
        gat_finalize<<<blocks, 256, 0, stream>>>(num, s, bias, total);
    }
}